// AdaptiveRefiner_87711822119203
// MI455X (gfx1250) — compile-verified
//
#include <hip/hip_runtime.h>
#include <hip/hip_bf16.h>
#include <math.h>

// Gaussian density map via separable-kernel batched GEMM on fp32 WMMA (gfx1250).
// out[b,y,x] = sum_n lutn[y - offr[n]] * lutn[x - offc[n]]
//   offr/offc = trunc(label - 46.5)  (Python int() truncation)
//   lutn[i]   = exp(-(i-46)^2/(2 s^2)) / sum  over i in [0,93)
// Formulated as per-image GEMM: M=N=512, K=256, fp32 WMMA 16x16x4.

typedef __attribute__((ext_vector_type(2))) float v2f;
typedef __attribute__((ext_vector_type(8))) float v8f;

#define KS       93
#define BIAS     465          // idx = y - off in [-465, 557] -> biased into [0, 1023]
#define LUTSZ    1024
#define NPTS     256
#define HW       512
#define BLK_M    64
#define BLK_N    64
#define KC       64           // K chunk staged in LDS
#define AST      68           // padded LDS stride: bank = (4*row + k) & 63, conflict-free
#define NTHREADS 128          // 4 waves (wave32)

__global__ __launch_bounds__(NTHREADS)
void gauss_density_wmma(const float* __restrict__ labels,  // [B, NPTS, 2]
                        const float* __restrict__ sigma_p, // [1]
                        float* __restrict__ out)           // [B, 1, HW, HW]
{
    __shared__ __align__(16) float s_lut[LUTSZ];     // zero-padded gaussian
    __shared__ int   s_offr[NPTS];
    __shared__ int   s_offc[NPTS];
    __shared__ __align__(16) float s_A[BLK_M * AST]; // A chunk: [row][k], stride AST
    __shared__ __align__(16) float s_B[BLK_N * AST]; // B chunk: [col][k], stride AST
    __shared__ float s_invsum;

    const int tid   = threadIdx.x;
    const int b     = blockIdx.z;
    const int tileY = blockIdx.y * BLK_M;
    const int tileX = blockIdx.x * BLK_N;

    // ---- zero-fill padded LUT (uniform: 1024/128 = 8 stores/thread) ----
#pragma unroll
    for (int i = 0; i < LUTSZ / NTHREADS; ++i)
        s_lut[tid + i * NTHREADS] = 0.0f;
    __syncthreads();

    // ---- unnormalized 1-D gaussian into the biased window ----
    {
        float sg = fabsf(sigma_p[0]);
        float inv2s2 = 1.0f / (2.0f * sg * sg);
        if (tid < KS) {
            float ax = (float)(tid - 46);
            s_lut[BIAS + tid] = __expf(-(ax * ax) * inv2s2);
        }
    }
    __syncthreads();
    if (tid == 0) {
        float s = 0.0f;
        for (int i = 0; i < KS; ++i) s += s_lut[BIAS + i];
        s_invsum = 1.0f / s;    // 2-D normalizer s^2 split across row/col factors
    }
    __syncthreads();
    if (tid < KS) s_lut[BIAS + tid] *= s_invsum;

    // ---- per-point footprint offsets ----
    {
        const float* lab = labels + (size_t)b * NPTS * 2;
        for (int t = tid; t < NPTS; t += NTHREADS) {
            s_offr[t] = (int)truncf(lab[2 * t + 0] - 46.5f); // label[...,0] -> row
            s_offc[t] = (int)truncf(lab[2 * t + 1] - 46.5f); // label[...,1] -> col
        }
    }
    __syncthreads();

    // ---- GEMM: 4 waves, each computes 16(M) x 64(N); K in LDS-staged chunks ----
    const int wave = tid >> 5;
    const int lane = tid & 31;
    const int m16  = lane & 15;
    const int kh   = lane >> 4;          // ISA A/B frag layout: half-wave K pairs

    const int kkStage = tid & 63;        // K column this thread stages
    const int rBase   = (tid >> 6) * 32; // rows/cols 0..31 or 32..63

    v8f acc[4];
    acc[0] = (v8f){}; acc[1] = (v8f){}; acc[2] = (v8f){}; acc[3] = (v8f){};

    for (int kb = 0; kb < NPTS; kb += KC) {
        // stage A[r][kk] and B[c][kk] for this K chunk (branch-free padded-LUT gathers)
        const int offR = s_offr[kb + kkStage];
        const int offC = s_offc[kb + kkStage];
#pragma unroll
        for (int i = 0; i < 32; ++i) {
            const int r = rBase + i;
            s_A[r * AST + kkStage] = s_lut[(tileY + r) - offR + BIAS];
            s_B[r * AST + kkStage] = s_lut[(tileX + r) - offC + BIAS];
        }
        __syncthreads();

        // compute: 16 k-steps of 4, fully unrolled; 5 aligned b64 LDS loads + 4 WMMA each
        const float* aRow = &s_A[(wave * 16 + m16) * AST];
#pragma unroll
        for (int kk = 0; kk < KC; kk += 4) {
            const int k0 = kk + 2 * kh;                 // even -> 8B-aligned v2f loads
            const v2f a = *(const v2f*)(aRow + k0);
#pragma unroll
            for (int j = 0; j < 4; ++j) {
                const v2f bf = *(const v2f*)(&s_B[(j * 16 + m16) * AST + k0]);
                acc[j] = __builtin_amdgcn_wmma_f32_16x16x4_f32(
                    false, a, false, bf, (short)0, acc[j], false, false);
            }
        }
        __syncthreads();
    }

    // ---- store: C/D layout — lanes 0-15: M=r, N=lane; lanes 16-31: M=r+8, N=lane-16
    float* outb = out + (size_t)b * HW * HW;
#pragma unroll
    for (int j = 0; j < 4; ++j) {
        const int col = tileX + j * 16 + m16;
#pragma unroll
        for (int r = 0; r < 8; ++r) {
            const int row = tileY + wave * 16 + r + kh * 8;
            outb[(size_t)row * HW + col] = acc[j][r];
        }
    }
}

extern "C" void kernel_launch(void* const* d_in, const int* in_sizes, int n_in,
                              void* d_out, int out_size, void* d_ws, size_t ws_size,
                              hipStream_t stream) {
    // inputs: d_in[0] = batch_images (shape-only, unused), d_in[1] = labels [B,256,2],
    //         d_in[2] = sigma [1]
    const float* labels = (const float*)d_in[1];
    const float* sigma  = (const float*)d_in[2];
    float* out = (float*)d_out;

    const int B = in_sizes[0] / (HW * HW);   // images are [B,1,512,512]
    dim3 grid(HW / BLK_N, HW / BLK_M, B);    // 8 x 8 x B
    gauss_density_wmma<<<grid, dim3(NTHREADS), 0, stream>>>(labels, sigma, out);
}